// GeneTokenizer_27084063769195
// MI455X (gfx1250) — compile-verified
//
#include <hip/hip_runtime.h>
#include <hip/hip_bf16.h>
#include <cstdint>
#include <cstddef>

// Soft descending-argsort permutation (Geneformer-style rank tokenizer), n = 8192.
//
// Pass 1 (rank_kernel):   ranks[j] = sum_i sigmoid((v_j - v_i)/T) - 0.5
//   The sum over i is folded into V_WMMA_F32_16X16X4_F32: each wave owns 16
//   columns (j), computes sigmoid values directly in the WMMA B-matrix (4x16
//   fp32) lane layout, and multiplies by an all-ones A-matrix (16x4) so the
//   matrix pipe performs the reduction while VALU/trans does the sigmoids.
// Pass 2 (softmax_kernel): row softmax of -(ranks[j] - (n-1-i))^2 / T,
//   store-bandwidth bound (256 MB out), one block per row, ranks in LDS,
//   logits kept in registers so exp() is evaluated once per element.

#define N_GENES 8192

typedef __attribute__((ext_vector_type(2))) float v2f;
typedef __attribute__((ext_vector_type(8))) float v8f;

__global__ __launch_bounds__(128) void rank_kernel(const float* __restrict__ expr,
                                                   const float* __restrict__ temp,
                                                   float* __restrict__ ranks) {
    __shared__ float s_expr[N_GENES];                       // 32 KB of 320 KB/WGP
    const int tid = threadIdx.x;

    // Cooperative load of the whole expression vector (8192 floats, b128 loads).
    {
        const float4* g = (const float4*)expr;
        float4* s = (float4*)s_expr;
#pragma unroll
        for (int c = 0; c < 16; ++c) s[c * 128 + tid] = g[c * 128 + tid];
    }
    __syncthreads();

    const float invT = __builtin_amdgcn_rcpf(temp[0]);

    const int lane  = tid & 31;
    const int wave  = tid >> 5;
    const int jbase = (blockIdx.x * 4 + wave) * 16;         // 16 columns per wave
    const int nCol  = lane & 15;                            // B-matrix N index
    const int hi    = lane >> 4;                            // lane-half selects K pair

    const float vj = s_expr[jbase + nCol];

    v8f acc = {};                                           // 16x16 f32 accumulator
    v2f ones;                                               // A = ones(16x4): both
    ones.x = 1.0f;                                          // VGPRs hold 1.0 in all lanes
    ones.y = 1.0f;

    // B-matrix 4x16 fp32 layout (wave32): VGPR0 -> K = 2*hi, VGPR1 -> K = 2*hi+1,
    // N = lane & 15. Each lane computes its two sigmoids straight into that slot.
    for (int ibase = 0; ibase < N_GENES; ibase += 4) {
        const int i0 = ibase + 2 * hi;
        const float x0 = (vj - s_expr[i0])     * invT;
        const float x1 = (vj - s_expr[i0 + 1]) * invT;
        v2f b;
        b.x = __builtin_amdgcn_rcpf(1.0f + __expf(-x0));    // sigmoid(x0)
        b.y = __builtin_amdgcn_rcpf(1.0f + __expf(-x1));    // sigmoid(x1)
        // D = ones(16x4) * B(4x16) + D : every row of D holds the column sums.
        acc = __builtin_amdgcn_wmma_f32_16x16x4_f32(
            /*neg_a=*/false, ones, /*neg_b=*/false, b,
            /*c_mod=*/(short)0, acc, /*reuse_a=*/false, /*reuse_b=*/false);
    }

    // D VGPR0: lanes 0-15 hold (M=0, N=lane) -> the column sum for N.
    if (lane < 16) ranks[jbase + nCol] = acc[0] - 0.5f;
}

__global__ __launch_bounds__(256) void softmax_kernel(const float* __restrict__ ranks,
                                                      const float* __restrict__ temp,
                                                      float* __restrict__ out) {
    __shared__ float s_ranks[N_GENES];                      // 32 KB
    __shared__ float s_red[8];                              // per-wave partials (8 waves)
    const int tid = threadIdx.x;
    const int row = blockIdx.x;

    {
        const float4* g = (const float4*)ranks;
        float4* s = (float4*)s_ranks;
#pragma unroll
        for (int c = 0; c < 8; ++c) s[c * 256 + tid] = g[c * 256 + tid];
    }
    __syncthreads();

    const float invT = __builtin_amdgcn_rcpf(temp[0]);
    const float tgt  = (float)(N_GENES - 1 - row);

    // Each thread owns 8 float4 chunks (32 columns), coalesced across the block.
    float lg[32];
    float mx = -3.402823466e38f;
    const float4* sr4 = (const float4*)s_ranks;
#pragma unroll
    for (int c = 0; c < 8; ++c) {
        const float4 r = sr4[c * 256 + tid];
        const float d0 = r.x - tgt, d1 = r.y - tgt, d2 = r.z - tgt, d3 = r.w - tgt;
        const float l0 = -d0 * d0 * invT, l1 = -d1 * d1 * invT;
        const float l2 = -d2 * d2 * invT, l3 = -d3 * d3 * invT;
        lg[c * 4 + 0] = l0; lg[c * 4 + 1] = l1;
        lg[c * 4 + 2] = l2; lg[c * 4 + 3] = l3;
        mx = fmaxf(mx, fmaxf(fmaxf(l0, l1), fmaxf(l2, l3)));
    }

    // Wave32 max reduction, then cross-wave via LDS.
#pragma unroll
    for (int off = 16; off > 0; off >>= 1) mx = fmaxf(mx, __shfl_xor(mx, off, 32));
    if ((tid & 31) == 0) s_red[tid >> 5] = mx;
    __syncthreads();
    mx = s_red[0];
#pragma unroll
    for (int w = 1; w < 8; ++w) mx = fmaxf(mx, s_red[w]);

    // exp once per element (kept in registers), sum-reduce.
    float sum = 0.0f;
#pragma unroll
    for (int k = 0; k < 32; ++k) {
        const float e = __expf(lg[k] - mx);
        lg[k] = e;
        sum += e;
    }
#pragma unroll
    for (int off = 16; off > 0; off >>= 1) sum += __shfl_xor(sum, off, 32);
    __syncthreads();                                        // s_red reuse guard
    if ((tid & 31) == 0) s_red[tid >> 5] = sum;
    __syncthreads();
    sum = 0.0f;
#pragma unroll
    for (int w = 0; w < 8; ++w) sum += s_red[w];
    const float rs = __builtin_amdgcn_rcpf(sum);

    // Coalesced b128 stores: 256 MB total across the grid -> ~11 us HBM floor.
    float4* o4 = (float4*)(out + (size_t)row * N_GENES);
#pragma unroll
    for (int c = 0; c < 8; ++c) {
        float4 v;
        v.x = lg[c * 4 + 0] * rs;
        v.y = lg[c * 4 + 1] * rs;
        v.z = lg[c * 4 + 2] * rs;
        v.w = lg[c * 4 + 3] * rs;
        o4[c * 256 + tid] = v;
    }
}

extern "C" void kernel_launch(void* const* d_in, const int* in_sizes, int n_in,
                              void* d_out, int out_size, void* d_ws, size_t ws_size,
                              hipStream_t stream) {
    (void)in_sizes; (void)n_in; (void)out_size; (void)ws_size;
    const float* expr = (const float*)d_in[0];
    const float* temp = (const float*)d_in[1];   // scalar temperature
    float* out   = (float*)d_out;
    float* ranks = (float*)d_ws;                 // 8192 floats of scratch

    // 128 blocks x 4 waves x 16 columns = 8192 ranks.
    rank_kernel<<<N_GENES / 64, 128, 0, stream>>>(expr, temp, ranks);
    // One block per output row.
    softmax_kernel<<<N_GENES, 256, 0, stream>>>(ranks, temp, out);
}